// MultiSimilarityLoss_37838661878693
// MI455X (gfx1250) — compile-verified
//
#include <hip/hip_runtime.h>
#include <math.h>

// MultiSimilarityLoss for MI455X (gfx1250), wave32, WMMA f32 16x16x4.
// B=4096 samples, D=1024 dims. sim = normalize(feats) @ normalize(feats)^T.
// Two WMMA GEMM passes (stats, then mined loss) — X (16MB f32) is L2-resident
// (192MB L2), so recomputing the GEMM is cheaper than materializing 64MB sim.
// Round 2: double-buffered K pipeline (overlap global b64 loads with wmma),
// A-strip staged in LDS with bank-conflict-free padding.

#define B_N 4096
#define D_K 1024
#define A_STRIDE 1028  // floats; 1028 % 64 == 4 -> lane m,half h reads banks
                       // {4m+2h, 4m+2h+1}: 32 lanes cover 64 distinct banks.

constexpr float ALPHA  = 2.0f;
constexpr float BETA   = 50.0f;
constexpr float MARGIN = 0.1f;
constexpr float THRESH = 0.5f;
constexpr float EPSV   = 1e-5f;

typedef float v2f __attribute__((ext_vector_type(2)));
typedef float v8f __attribute__((ext_vector_type(8)));

// Monotonic float <-> uint mapping so unsigned atomicMax/Min give
// order-independent (deterministic) float max/min.
__device__ __forceinline__ unsigned mapf(float f) {
  unsigned b = __float_as_uint(f);
  return (b & 0x80000000u) ? ~b : (b | 0x80000000u);
}
__device__ __forceinline__ float unmapf(unsigned u) {
  return (u & 0x80000000u) ? __uint_as_float(u & 0x7FFFFFFFu)
                           : __uint_as_float(~u);
}
// mapf(-inf) = 0x007FFFFF ; mapf(+inf) = 0xFF800000
#define MAP_NEG_INF 0x007FFFFFu
#define MAP_POS_INF 0xFF800000u

__global__ void ms_init(unsigned* __restrict__ mn, unsigned* __restrict__ mp) {
  int i = blockIdx.x * blockDim.x + threadIdx.x;
  if (i < B_N) { mn[i] = 0u; mp[i] = 0xFFFFFFFFu; }
}

// One block per row: L2-normalize, write f32 X to workspace.
__global__ void __launch_bounds__(256) ms_normalize(const float* __restrict__ feats,
                                                    float* __restrict__ xn) {
  __shared__ float red[256];
  const int row = blockIdx.x;
  const float* fr = feats + (size_t)row * D_K;
  float ss = 0.0f;
#pragma unroll
  for (int j = 0; j < 4; ++j) {
    float v = fr[threadIdx.x + 256 * j];
    ss += v * v;
  }
  red[threadIdx.x] = ss;
  __syncthreads();
  for (int s = 128; s > 0; s >>= 1) {            // fixed tree -> deterministic
    if (threadIdx.x < s) red[threadIdx.x] += red[threadIdx.x + s];
    __syncthreads();
  }
  float inv = 1.0f / fmaxf(sqrtf(red[0]), 1e-12f);
  float* xr = xn + (size_t)row * D_K;
#pragma unroll
  for (int j = 0; j < 4; ++j) {
    int k = threadIdx.x + 256 * j;
    xr[k] = fr[k] * inv;
  }
}

// One K chunk = 4 wmma k-steps (16 K values): A frags from LDS, B from global.
__device__ __forceinline__ void load_chunk(const float* aL,
                                           const float* const bptr[4], int koff,
                                           v2f a[4], v2f b[4][4]) {
#pragma unroll
  for (int q = 0; q < 4; ++q) {
    a[q] = *(const v2f*)(aL + koff + 4 * q);
#pragma unroll
    for (int t = 0; t < 4; ++t)
      b[q][t] = *(const v2f*)(bptr[t] + koff + 4 * q);
  }
}

__device__ __forceinline__ void wmma_chunk(const v2f a[4], const v2f b[4][4],
                                           v8f acc[4]) {
#pragma unroll
  for (int q = 0; q < 4; ++q) {
#pragma unroll
    for (int t = 0; t < 4; ++t) {
      acc[t] = __builtin_amdgcn_wmma_f32_16x16x4_f32(
          /*neg_a=*/false, a[q], /*neg_b=*/false, b[q][t],
          /*c_mod=*/(short)0, acc[t], /*reuse_a=*/false, /*reuse_b=*/false);
    }
  }
}

// PASS 1: per-row max_neg / min_pos via mapped-uint atomics.
// PASS 2: mined pos/neg exp-sums, per-16-row-strip loss written to strip_loss.
template <int PASS>
__global__ void __launch_bounds__(256) ms_gemm(const float* __restrict__ xn,
                                               const int* __restrict__ labels,
                                               unsigned* __restrict__ mn,
                                               unsigned* __restrict__ mp,
                                               float* __restrict__ strip_loss) {
  __shared__ float As[16 * A_STRIDE];   // 64KB+pad A strip (of 320KB WGP LDS)
  __shared__ float wPos[8][16];
  __shared__ float wNeg[8][16];
  __shared__ float rowLoss[16];

  const int wave  = threadIdx.x >> 5;   // 8 waves of 32
  const int lane  = threadIdx.x & 31;
  const int m     = lane & 15;          // M (A) / N (B,C) index within tile
  const int khalf = lane >> 4;          // which K half-pair this lane owns
  const int rowBase = blockIdx.x * 16;

  // Cooperative stage of the block's 16-row A strip into LDS (coalesced).
  for (int i = threadIdx.x; i < 16 * D_K; i += 256) {
    int r = i >> 10, k = i & (D_K - 1);
    As[r * A_STRIDE + k] = xn[(size_t)(rowBase + r) * D_K + k];
  }
  __syncthreads();

  // A fragment source: As[m][k + 2*khalf .. +1]  (ISA f32 A layout)
  const float* aL = As + m * A_STRIDE + 2 * khalf;

  int lrow[8];
#pragma unroll
  for (int r = 0; r < 8; ++r) lrow[r] = labels[rowBase + r + 8 * khalf];

  float runMax[8], runMin[8];                 // pass 1
  float maxNeg[8], minPos[8];                 // pass 2
  bool  hasNeg[8], hasPos[8];
  float posAcc[8], negAcc[8];
#pragma unroll
  for (int r = 0; r < 8; ++r) {
    if (PASS == 1) {
      runMax[r] = -INFINITY;
      runMin[r] =  INFINITY;
    } else {
      int row = rowBase + r + 8 * khalf;
      unsigned un = mn[row], up = mp[row];
      hasNeg[r] = un > MAP_NEG_INF;
      hasPos[r] = up < MAP_POS_INF;
      maxNeg[r] = hasNeg[r] ? unmapf(un) : -INFINITY;
      minPos[r] = hasPos[r] ? unmapf(up) :  INFINITY;
      posAcc[r] = 0.0f;
      negAcc[r] = 0.0f;
    }
  }

  // 256 column tiles = 8 groups x 8 waves x 4 accumulators.
  for (int g = 0; g < 8; ++g) {
    const int ct0 = (g * 8 + wave) * 4;
    const float* bptr[4];
    int lcol[4];
#pragma unroll
    for (int t = 0; t < 4; ++t) {
      int colBase = (ct0 + t) * 16;
      // B fragment source: X[colBase+m][k + 2*khalf .. +1]
      bptr[t] = xn + (size_t)(colBase + m) * D_K + 2 * khalf;
      lcol[t] = labels[colBase + m];
    }

    v8f acc[4];
#pragma unroll
    for (int t = 0; t < 4; ++t) acc[t] = (v8f){0, 0, 0, 0, 0, 0, 0, 0};

    // Double-buffered K pipeline: chunk c+1's loads fly while chunk c's
    // wmmas execute -> no full s_wait_loadcnt 0 in front of each wmma.
    v2f aA[4], bA[4][4], aB[4], bB[4][4];
    load_chunk(aL, bptr, 0, aA, bA);
#pragma unroll 1
    for (int kk = 0; kk + 32 <= D_K - 16; kk += 32) {
      load_chunk(aL, bptr, kk + 16, aB, bB);
      wmma_chunk(aA, bA, acc);
      load_chunk(aL, bptr, kk + 32, aA, bA);
      wmma_chunk(aB, bB, acc);
    }
    load_chunk(aL, bptr, D_K - 16, aB, bB);
    wmma_chunk(aA, bA, acc);
    wmma_chunk(aB, bB, acc);

    // Tile epilogue. C layout: VGPR r -> M = r + 8*khalf, N = lane&15.
#pragma unroll
    for (int t = 0; t < 4; ++t) {
      int lc = lcol[t];
#pragma unroll
      for (int r = 0; r < 8; ++r) {
        float v = acc[t][r];
        bool same = (lrow[r] == lc);
        if (PASS == 1) {
          if (!same)                      runMax[r] = fmaxf(runMax[r], v);
          if (same && v < 1.0f - EPSV)    runMin[r] = fminf(runMin[r], v);
        } else {
          if (same && v < 1.0f - EPSV && hasNeg[r] && (v - MARGIN < maxNeg[r]))
            posAcc[r] += expf(-ALPHA * (v - THRESH));
          if (!same && hasPos[r] && (v + MARGIN > minPos[r]))
            negAcc[r] += expf(BETA * (v - THRESH));
        }
      }
    }
  }

  if (PASS == 1) {
    // Reduce over the 16-lane N dimension (xor <= 8 stays within halves).
#pragma unroll
    for (int r = 0; r < 8; ++r) {
#pragma unroll
      for (int s = 1; s < 16; s <<= 1) {
        runMax[r] = fmaxf(runMax[r], __shfl_xor(runMax[r], s));
        runMin[r] = fminf(runMin[r], __shfl_xor(runMin[r], s));
      }
    }
    if (m == 0) {
#pragma unroll
      for (int r = 0; r < 8; ++r) {
        int row = rowBase + r + 8 * khalf;
        if (runMax[r] > -INFINITY) atomicMax(&mn[row], mapf(runMax[r]));
        if (runMin[r] <  INFINITY) atomicMin(&mp[row], mapf(runMin[r]));
      }
    }
  } else {
#pragma unroll
    for (int r = 0; r < 8; ++r) {
#pragma unroll
      for (int s = 1; s < 16; s <<= 1) {
        posAcc[r] += __shfl_xor(posAcc[r], s);
        negAcc[r] += __shfl_xor(negAcc[r], s);
      }
    }
    if (m == 0) {
#pragma unroll
      for (int r = 0; r < 8; ++r) {
        wPos[wave][r + 8 * khalf] = posAcc[r];
        wNeg[wave][r + 8 * khalf] = negAcc[r];
      }
    }
    __syncthreads();
    if (threadIdx.x < 16) {
      float ps = 0.0f, ns = 0.0f;
      for (int w = 0; w < 8; ++w) {      // fixed order -> deterministic
        ps += wPos[w][threadIdx.x];
        ns += wNeg[w][threadIdx.x];
      }
      float l = 0.0f;
      if (ps > 0.0f) l += log1pf(ps) / ALPHA;
      if (ns > 0.0f) l += log1pf(ns) / BETA;
      rowLoss[threadIdx.x] = l;
    }
    __syncthreads();
    if (threadIdx.x == 0) {
      float s = 0.0f;
      for (int i = 0; i < 16; ++i) s += rowLoss[i];
      strip_loss[blockIdx.x] = s;
    }
  }
}

__global__ void __launch_bounds__(256) ms_finalize(const float* __restrict__ strip_loss,
                                                   float* __restrict__ out) {
  __shared__ float red[256];
  red[threadIdx.x] = strip_loss[threadIdx.x];
  __syncthreads();
  for (int s = 128; s > 0; s >>= 1) {
    if (threadIdx.x < s) red[threadIdx.x] += red[threadIdx.x + s];
    __syncthreads();
  }
  if (threadIdx.x == 0) out[0] = red[0] / (float)B_N;
}

extern "C" void kernel_launch(void* const* d_in, const int* in_sizes, int n_in,
                              void* d_out, int out_size, void* d_ws, size_t ws_size,
                              hipStream_t stream) {
  const float* feats  = (const float*)d_in[0];
  const int*   labels = (const int*)d_in[1];

  char* ws = (char*)d_ws;
  float*    xn    = (float*)ws;                                 // 16 MB
  unsigned* mn    = (unsigned*)(ws + (size_t)B_N * D_K * 4);    // 16 KB
  unsigned* mp    = mn + B_N;                                   // 16 KB
  float*    strip = (float*)(mp + B_N);                         // 1 KB

  ms_init<<<(B_N + 255) / 256, 256, 0, stream>>>(mn, mp);
  ms_normalize<<<B_N, 256, 0, stream>>>(feats, xn);
  ms_gemm<1><<<B_N / 16, 256, 0, stream>>>(xn, labels, mn, mp, strip);
  ms_gemm<2><<<B_N / 16, 256, 0, stream>>>(xn, labels, mn, mp, strip);
  ms_finalize<<<1, 256, 0, stream>>>(strip, (float*)d_out);
}